// DOSLoss_49005576847592
// MI455X (gfx1250) — compile-verified
//
#include <hip/hip_runtime.h>
#include <hip/hip_bf16.h>
#include <math.h>

// Problem sizes (fixed by the reference).
constexpr int kD = 2048;   // feature dim
constexpr int kK = 4096;   // neighbors
constexpr int kW = 4096;   // rows of w
constexpr int kC = 1000;   // classes

typedef __attribute__((ext_vector_type(2))) float v2f;
typedef __attribute__((ext_vector_type(4))) float v4f;
typedef __attribute__((ext_vector_type(8))) float v8f;

// ---------------------------------------------------------------------------
// Kernel 1: d[r] = || f - n[r] ||_2 via V_WMMA_F32_16X16X4_F32.
// Block = 256 threads = 8 waves; block b owns rows 16b..16b+15.
// Wave w covers k in [256w, 256w+256): 64 WMMA chunks of K=4.
// A tile (16x4 f32, ISA layout): lane m<16 -> row m, K0/K1; lane m+16 -> row m, K2/K3.
// Passing the same regs as A and B gives C = X * X^T; diagonal = partial sum(diff^2).
// ---------------------------------------------------------------------------
__global__ __launch_bounds__(256) void k_dist(const float* __restrict__ f,
                                              const float* __restrict__ n,
                                              float* __restrict__ d_arr) {
  __shared__ float partial[8][16];
  const int tid  = threadIdx.x;
  const int wave = tid >> 5;
  const int lane = tid & 31;
  const int rowTile = blockIdx.x;            // 0..255
  const int m    = lane & 15;                // row within tile
  const int kOff = (lane >> 4) << 1;         // 0 for K0/K1, 2 for K2/K3

  const float* rowPtr = n + (size_t)(rowTile * 16 + m) * kD + kOff;
  const float* fPtr   = f + kOff;
  const int kStart = wave * 256;

  v8f acc = {};
  #pragma unroll 4
  for (int c = 0; c < 64; ++c) {
    const int k = kStart + c * 4;
    const v2f nv = *(const v2f*)(rowPtr + k);   // 8B aligned
    const v2f fv = *(const v2f*)(fPtr + k);
    const v2f a  = nv - fv;                     // diff tile chunk
    // D = A * A^T + C  (f32, round-to-nearest-even)
    acc = __builtin_amdgcn_wmma_f32_16x16x4_f32(
        /*neg_a=*/false, a, /*neg_b=*/false, a,
        /*c_mod=*/(short)0, acc, /*reuse_a=*/false, /*reuse_b=*/false);
  }

  // Diagonal extraction (C/D layout): lane L<8 holds C[L,L] in acc[L];
  // lane L in 24..31 holds C[L-16,L-16] in acc[L-24].  In both cases idx = L&7.
  float v = acc[0];
  #pragma unroll
  for (int i = 1; i < 8; ++i)
    if ((lane & 7) == i) v = acc[i];

  if (lane < 8)        partial[wave][lane]      = v;
  else if (lane >= 24) partial[wave][lane - 16] = v;
  __syncthreads();

  if (tid < 16) {
    float s = 0.f;
    #pragma unroll
    for (int wv = 0; wv < 8; ++wv) s += partial[wv][tid];
    d_arr[rowTile * 16 + tid] = sqrtf(s);
  }
}

// ---------------------------------------------------------------------------
// Kernel 2: ce[r] = logsumexp(cls_score[r,:]) - cls_score[r,target]
// One block (128 threads) per row; row cached in registers via b128 loads.
// 1000 floats = 250 float4 (row stride 4000 B is 16 B aligned).
// Out-of-range lanes hold -INF: max ignores them, exp(-INF - m) == 0.
// ---------------------------------------------------------------------------
__global__ __launch_bounds__(128) void k_ce(const float* __restrict__ cls,
                                            const long long* __restrict__ tgt,
                                            float* __restrict__ ce) {
  __shared__ float red[128];
  const int row = blockIdx.x;
  const int tid = threadIdx.x;
  const float* rp = cls + (size_t)row * kC;
  const v4f* rp4  = (const v4f*)rp;

  float s[8];
  float mx = -INFINITY;
  #pragma unroll
  for (int j = 0; j < 2; ++j) {
    const int idx4 = j * 128 + tid;           // 0..255, valid < 250
    v4f v;
    if (idx4 < kC / 4) v = rp4[idx4];
    else               v = (v4f){-INFINITY, -INFINITY, -INFINITY, -INFINITY};
    #pragma unroll
    for (int e = 0; e < 4; ++e) {
      s[j * 4 + e] = v[e];
      mx = fmaxf(mx, v[e]);
    }
  }
  red[tid] = mx; __syncthreads();
  for (int off = 64; off > 0; off >>= 1) {
    if (tid < off) red[tid] = fmaxf(red[tid], red[tid + off]);
    __syncthreads();
  }
  mx = red[0]; __syncthreads();

  float z = 0.f;
  #pragma unroll
  for (int j = 0; j < 8; ++j) z += expf(s[j] - mx);   // -INF slots contribute 0
  red[tid] = z; __syncthreads();
  for (int off = 64; off > 0; off >>= 1) {
    if (tid < off) red[tid] += red[tid + off];
    __syncthreads();
  }
  if (tid == 0) {
    const int t = (int)tgt[0];
    ce[row] = logf(red[0]) + mx - rp[t];
  }
}

// ---------------------------------------------------------------------------
// Kernel 3: per row of w:  t = w_row * d  (registers, 16/thread, b128 loads),
//   frow = sum(t);  m = max(-t) = -min(t);
//   Z = sum exp(tmin - t);  S = sum exp(tmin - t) * ce;  grow = S / Z.
// One block (256 threads) per row; w streamed once (64 MB total, the hot path).
// ---------------------------------------------------------------------------
__global__ __launch_bounds__(256) void k_wrow(const float* __restrict__ w,
                                              const float* __restrict__ d_arr,
                                              const float* __restrict__ ce,
                                              float* __restrict__ grow,
                                              float* __restrict__ frow) {
  __shared__ float red[256];
  const int row = blockIdx.x;
  const int tid = threadIdx.x;
  const v4f* wr4 = (const v4f*)(w + (size_t)row * kK);  // 16 KB-aligned row
  const v4f* d4  = (const v4f*)d_arr;                   // L2-resident, 16 KB
  const v4f* c4  = (const v4f*)ce;

  float t[16], cv[16];
  float fsum = 0.f, tmin = INFINITY;
  #pragma unroll
  for (int j = 0; j < 4; ++j) {
    const int idx4 = j * 256 + tid;           // coalesced b128
    const v4f wv = wr4[idx4];
    const v4f dv = d4[idx4];
    const v4f cev = c4[idx4];
    #pragma unroll
    for (int e = 0; e < 4; ++e) {
      const float tv = wv[e] * dv[e];
      t[j * 4 + e]  = tv;
      cv[j * 4 + e] = cev[e];
      fsum += tv;
      tmin = fminf(tmin, tv);
    }
  }

  // block min(t)
  red[tid] = tmin; __syncthreads();
  for (int off = 128; off > 0; off >>= 1) {
    if (tid < off) red[tid] = fminf(red[tid], red[tid + off]);
    __syncthreads();
  }
  tmin = red[0]; __syncthreads();

  float z = 0.f, sc = 0.f;
  #pragma unroll
  for (int j = 0; j < 16; ++j) {
    const float e = expf(tmin - t[j]);        // exp(-t - m), m = -tmin
    z  += e;
    sc += e * cv[j];
  }

  red[tid] = fsum; __syncthreads();
  for (int off = 128; off > 0; off >>= 1) {
    if (tid < off) red[tid] += red[tid + off];
    __syncthreads();
  }
  const float FS = red[0]; __syncthreads();

  red[tid] = z; __syncthreads();
  for (int off = 128; off > 0; off >>= 1) {
    if (tid < off) red[tid] += red[tid + off];
    __syncthreads();
  }
  const float Z = red[0]; __syncthreads();

  red[tid] = sc; __syncthreads();
  for (int off = 128; off > 0; off >>= 1) {
    if (tid < off) red[tid] += red[tid + off];
    __syncthreads();
  }
  if (tid == 0) {
    grow[row] = red[0] / Z;
    frow[row] = FS;
  }
}

// ---------------------------------------------------------------------------
// Kernel 4: deterministic final reduction: out = sum(grow) + sum(frow)
// ---------------------------------------------------------------------------
__global__ __launch_bounds__(256) void k_final(const float* __restrict__ grow,
                                               const float* __restrict__ frow,
                                               float* __restrict__ out) {
  __shared__ float red[256];
  const int tid = threadIdx.x;
  float s = 0.f;
  for (int idx = tid; idx < kW; idx += 256) s += grow[idx] + frow[idx];
  red[tid] = s; __syncthreads();
  for (int off = 128; off > 0; off >>= 1) {
    if (tid < off) red[tid] += red[tid + off];
    __syncthreads();
  }
  if (tid == 0) out[0] = red[0];
}

// ---------------------------------------------------------------------------
extern "C" void kernel_launch(void* const* d_in, const int* in_sizes, int n_in,
                              void* d_out, int out_size, void* d_ws, size_t ws_size,
                              hipStream_t stream) {
  const float*     deep = (const float*)d_in[0];      // [1, 2048]
  const float*     cls  = (const float*)d_in[1];      // [4096, 1000]
  const long long* tgt  = (const long long*)d_in[2];  // [1] int64
  const float*     n    = (const float*)d_in[3];      // [4096, 2048]
  const float*     w    = (const float*)d_in[4];      // [4096, 4096]
  float* out = (float*)d_out;

  float* wsf  = (float*)d_ws;
  float* dv   = wsf;               // [4096]
  float* ce   = wsf + kK;          // [4096]
  float* grow = wsf + 2 * kK;      // [4096]
  float* frow = wsf + 3 * kK;      // [4096]

  k_dist <<<kK / 16, 256, 0, stream>>>(deep, n, dv);
  k_ce   <<<kK,      128, 0, stream>>>(cls, tgt, ce);
  k_wrow <<<kW,      256, 0, stream>>>(w, dv, ce, grow, frow);
  k_final<<<1,       256, 0, stream>>>(grow, frow, out);
}